// antisymgnn_26422638805509
// MI455X (gfx1250) — compile-verified
//
#include <hip/hip_runtime.h>
#include <hip/hip_bf16.h>

#define NN 100000
#define EE 1250000
#define CC 64
#define NUM_ITERS 4
#define GAMMA 0.1f
#define EPSILON 0.1f

typedef __attribute__((ext_vector_type(2))) float v2f;
typedef __attribute__((ext_vector_type(8))) float v8f;

// ---------------------------------------------------------------------------
// Kernel 0: A = W - W^T - gamma*I   (64x64, run once)
// ---------------------------------------------------------------------------
__global__ void build_A_kernel(const float* __restrict__ W, float* __restrict__ Amat) {
    for (int i = threadIdx.x; i < CC * CC; i += blockDim.x) {
        int j = i >> 6;       // row
        int k = i & 63;       // col
        float v = W[j * CC + k] - W[k * CC + j];
        if (j == k) v -= GAMMA;
        Amat[i] = v;
    }
}

// ---------------------------------------------------------------------------
// Kernel 1: fused dual GEMM via V_WMMA_F32_16X16X4_F32
//   neigh    = x @ lin_w^T           (no bias)
//   agg_base = x @ A^T + b           (seed for the scatter-add)
// One wave (32 lanes) owns a 16-row tile of x; 8 waves per block. Weights are
// staged into LDS with GLOBAL_LOAD_ASYNC_TO_LDS_B128 (ASYNCcnt-tracked DMA,
// no VGPR round trip), then consumed via ds loads as B fragments.
// ---------------------------------------------------------------------------
__global__ void __launch_bounds__(256)
gemm_dual_kernel(const float* __restrict__ x,
                 const float* __restrict__ linw,
                 const float* __restrict__ Amat,
                 const float* __restrict__ bias,
                 float* __restrict__ neigh,
                 float* __restrict__ aggbase,
                 int nTiles) {
    __shared__ float lw[CC * CC];   // lin_w row-major [j][k]
    __shared__ float aw[CC * CC];   // A     row-major [j][k]

    {
        // Flat-LDS address low 32 bits == byte offset within the group segment.
        unsigned lds_lw = (unsigned)(size_t)&lw[0];
        unsigned lds_aw = (unsigned)(size_t)&aw[0];
        unsigned long long g_lw = (unsigned long long)linw;
        unsigned long long g_aw = (unsigned long long)Amat;
        // 16 KB per matrix / 256 threads = 4 x 16B async copies per thread.
#pragma unroll
        for (int c = 0; c < 4; ++c) {
            unsigned off = (threadIdx.x + c * 256u) * 16u;
            asm volatile("global_load_async_to_lds_b128 %0, %1, %2"
                         :: "v"(lds_lw + off), "v"(off), "s"(g_lw)
                         : "memory");
            asm volatile("global_load_async_to_lds_b128 %0, %1, %2"
                         :: "v"(lds_aw + off), "v"(off), "s"(g_aw)
                         : "memory");
        }
        asm volatile("s_wait_asynccnt 0" ::: "memory");
    }
    __syncthreads();

    const int wave = threadIdx.x >> 5;
    const int lane = threadIdx.x & 31;
    const int tile = blockIdx.x * (blockDim.x >> 5) + wave;
    if (tile >= nTiles) return;

    // A-fragment layout (16x4 f32): lanes 0-15 -> M=lane, K = ks*4 + {0,1}
    //                               lanes16-31 -> M=lane-16, K = ks*4 + {2,3}
    const int mrow  = lane & 15;
    const int khalf = (lane >> 4) * 2;          // 0 or 2
    const int row   = tile * 16 + mrow;

    v2f afrag[16];
    const float* xrow = x + (size_t)row * CC;
#pragma unroll
    for (int ks = 0; ks < 16; ++ks)
        afrag[ks] = *(const v2f*)(xrow + ks * 4 + khalf);

    const int bcol  = lane & 15;                // N-column within 16-wide tile
    const int rbase = tile * 16 + (lane >> 4) * 8;

#pragma unroll
    for (int j = 0; j < 4; ++j) {               // four 16-wide output col tiles
        const int col0 = j * 16;

        v8f cn = {};                            // neigh accumulator (zero C)
        v8f ca;                                 // conv accumulator seeded w/ bias
        const float bv = bias[col0 + bcol];
#pragma unroll
        for (int v = 0; v < 8; ++v) ca[v] = bv;

        const float* lwrow = &lw[(col0 + bcol) * CC];
        const float* awrow = &aw[(col0 + bcol) * CC];
#pragma unroll
        for (int ks = 0; ks < 16; ++ks) {
            // B-fragment (4x16 f32): lanes 0-15 rows K={0,1}, lanes16-31 K={2,3}
            v2f bL = *(const v2f*)(lwrow + ks * 4 + khalf);
            cn = __builtin_amdgcn_wmma_f32_16x16x4_f32(
                false, afrag[ks], false, bL, (short)0, cn, false, false);
            v2f bA = *(const v2f*)(awrow + ks * 4 + khalf);
            ca = __builtin_amdgcn_wmma_f32_16x16x4_f32(
                false, afrag[ks], false, bA, (short)0, ca, false, false);
        }

        // C/D layout: VGPR v, lanes 0-15 -> M=v, lanes 16-31 -> M=v+8; col=lane&15
#pragma unroll
        for (int v = 0; v < 8; ++v) {
            neigh  [(size_t)(rbase + v) * CC + col0 + bcol] = cn[v];
            aggbase[(size_t)(rbase + v) * CC + col0 + bcol] = ca[v];
        }
    }
}

// ---------------------------------------------------------------------------
// Kernel 2: edge scatter.  16 lanes per edge, float4 gather of neigh[src],
// scale by edge_weight, 4x global_atomic_add_f32 into agg[dst].
// ---------------------------------------------------------------------------
__global__ void __launch_bounds__(256)
scatter_kernel(const float* __restrict__ neigh,
               const int* __restrict__ src,
               const int* __restrict__ dst,
               const float* __restrict__ ew,
               float* __restrict__ agg,
               int E) {
    const int tid = blockIdx.x * blockDim.x + threadIdx.x;
    const int e = tid >> 4;
    if (e >= E) return;
    const int q = tid & 15;

    const int s = src[e];
    const int d = dst[e];
    const float w = ew[e];

    const float4 v = *((const float4*)(neigh + (size_t)s * CC) + q);
    float* ap = agg + (size_t)d * CC + q * 4;
    atomicAdd(ap + 0, w * v.x);
    atomicAdd(ap + 1, w * v.y);
    atomicAdd(ap + 2, w * v.z);
    atomicAdd(ap + 3, w * v.w);
}

// ---------------------------------------------------------------------------
// Kernel 3: x_new = x + eps * tanh(agg)   (float4 vectorized)
// ---------------------------------------------------------------------------
__global__ void __launch_bounds__(256)
update_kernel(const float* __restrict__ xin,
              const float* __restrict__ agg,
              float* __restrict__ xout,
              int total4) {
    const int i = blockIdx.x * blockDim.x + threadIdx.x;
    if (i >= total4) return;
    float4 xv = ((const float4*)xin)[i];
    float4 av = ((const float4*)agg)[i];
    xv.x += EPSILON * tanhf(av.x);
    xv.y += EPSILON * tanhf(av.y);
    xv.z += EPSILON * tanhf(av.z);
    xv.w += EPSILON * tanhf(av.w);
    ((float4*)xout)[i] = xv;
}

// ---------------------------------------------------------------------------
extern "C" void kernel_launch(void* const* d_in, const int* in_sizes, int n_in,
                              void* d_out, int out_size, void* d_ws, size_t ws_size,
                              hipStream_t stream) {
    const float* x0   = (const float*)d_in[0];          // [N, C]
    const int*   ei   = (const int*)d_in[1];            // [2, E] flat
    const float* ew   = (const float*)d_in[2];          // [E]
    const float* W    = (const float*)d_in[3];          // [C, C]
    const float* linw = (const float*)d_in[4];          // [C, C]
    const float* bias = (const float*)d_in[5];          // [C]

    const int* src = ei;            // edge_index[0]
    const int* dst = ei + EE;       // edge_index[1]

    float* out = (float*)d_out;

    // Workspace layout: A(16KB) | neigh(N*C) | agg(N*C) | xalt(N*C)
    char* ws = (char*)d_ws;
    float* Amat  = (float*)ws;
    float* neigh = (float*)(ws + 16384);
    float* agg   = neigh + (size_t)NN * CC;
    float* xalt  = agg + (size_t)NN * CC;

    build_A_kernel<<<1, 256, 0, stream>>>(W, Amat);

    const int nTiles     = NN / 16;                       // 6250
    const int gemmBlocks = (nTiles + 7) / 8;              // 8 waves/block
    const int scatBlocks = ((EE * 16) + 255) / 256;
    const int total4     = (NN * CC) / 4;
    const int updBlocks  = (total4 + 255) / 256;

    // Ping-pong so iteration 4 lands in d_out:
    // x0 -> xalt -> out -> xalt -> out
    const float* xin = x0;
    float* xouts[NUM_ITERS] = { xalt, out, xalt, out };

    for (int it = 0; it < NUM_ITERS; ++it) {
        gemm_dual_kernel<<<gemmBlocks, 256, 0, stream>>>(
            xin, linw, Amat, bias, neigh, agg, nTiles);
        scatter_kernel<<<scatBlocks, 256, 0, stream>>>(
            neigh, src, dst, ew, agg, EE);
        update_kernel<<<updBlocks, 256, 0, stream>>>(
            xin, agg, xouts[it], total4);
        xin = xouts[it];
    }
}